// Plane_propagated_depth_guider_18537078849866
// MI455X (gfx1250) — compile-verified
//
#include <hip/hip_runtime.h>
#include <stdint.h>

#define HH 480
#define WW 640
#define NMASK 128
#define QQ 8192
#define KPTS 10
#define NP (HH * WW)            // 307200 pixels
#define SGCNT (NP / 64)         // 4800 super-groups of 64 pixels
#define MOM_BLOCKS 600          // 600 blocks * 4 waves = 2400 waves; 2 super-groups each
#define MOM_THREADS 128
#define ROWB 72                 // LDS row stride (64 mask bytes + pad) -> conflict-free u16 reads, 8B-aligned stores

typedef __attribute__((ext_vector_type(2))) float v2f;
typedef __attribute__((ext_vector_type(8))) float v8f;

// ---------------------------------------------------------------- init
__global__ void init_kernel(unsigned long long* pk, float* Sg) {
  int i = blockIdx.x * blockDim.x + threadIdx.x;
  if (i < NP) pk[i] = 0ull;
  if (i < NMASK * 16) Sg[i] = 0.0f;
}

// ---------------------------------------------------------------- moments via f32 WMMA
// S[128 x 16] = W[128 x P] * Qf[P x 16], W = (gt_mask & random_mask), Qf = valid * features
__global__ __launch_bounds__(MOM_THREADS)
void moment_kernel(const float* __restrict__ d0,
                   const unsigned char* __restrict__ gm,
                   const unsigned char* __restrict__ rm,
                   const float* __restrict__ gtd,
                   const float* __restrict__ kmat,
                   float* __restrict__ Sg) {
  __shared__ __align__(16) unsigned char smem[4 * 128 * ROWB];  // 36864 B

  const int tid  = threadIdx.x;
  const int wave = tid >> 5;
  const int lane = tid & 31;
  const int hi   = lane >> 4;     // 0: K=0,1  1: K=2,3
  const int col  = lane & 15;     // B column / A row-in-tile
  unsigned char* smw = smem + wave * (128 * ROWB);

  const float rfx = 1.0f / kmat[0];
  const float rfy = 1.0f / kmat[4];

  // column -> feature = (u * v) * valid, u,v in {px, py, 1, d, 0}
  const float su_px = (col==0||col==1||col==2||col==6) ? 1.f : 0.f;
  const float su_py = (col==3||col==4||col==7)         ? 1.f : 0.f;
  const float su_1  = (col==5||col==8)                 ? 1.f : 0.f;
  const float sv_px = (col==0)                         ? 1.f : 0.f;
  const float sv_py = (col==1||col==3)                 ? 1.f : 0.f;
  const float sv_1  = (col==2||col==4||col==5)         ? 1.f : 0.f;
  const float sv_d  = (col==6||col==7||col==8)         ? 1.f : 0.f;

  const v8f vzero = {0.f,0.f,0.f,0.f,0.f,0.f,0.f,0.f};
  v8f acc[8];
  #pragma unroll
  for (int t = 0; t < 8; ++t) acc[t] = vzero;

  const int gw = blockIdx.x * 4 + wave;            // 0..2399
  const int rbase = lane >> 3;                     // 0..3
  const int off   = (lane & 7) << 3;               // 0..56

  for (int si = 0; si < 2; ++si) {
    const int s  = gw * 2 + si;                    // 0..4799 (exact)
    const int PB = s * 64;                         // pixel base

    // ---- stage combined masks: 128 rows x 64 bytes, fully coalesced b64 loads
    #pragma unroll 4
    for (int it = 0; it < 32; ++it) {
      const int rrow = it * 4 + rbase;
      const size_t gi = (size_t)rrow * NP + (size_t)(PB + off);
      unsigned long long gv = *(const unsigned long long*)(gm + gi);
      unsigned long long rv = *(const unsigned long long*)(rm + gi);
      *(unsigned long long*)(smw + rrow * ROWB + off) = gv & rv;
    }
    __syncthreads();

    // ---- 16 K-groups of 4 pixels; 8 n-tiles (all 128 masks) per group
    for (int g = 0; g < 16; ++g) {
      const int pl = (g << 2) + (hi << 1);         // this lane's K pixel pair
      const int p  = PB + pl;
      const float2 dz = *(const float2*)(d0 + p);
      const float2 gz = *(const float2*)(gtd + p);

      const int r0 = p / WW;        const int c0 = p - r0 * WW;
      const int r1 = (p + 1) / WW;  const int c1 = (p + 1) - r1 * WW;
      const float ad0 = fabsf(dz.x), ad1 = fabsf(dz.y);
      const float px0 = ((float)c0 - 320.0f) * ad0 * rfx;
      const float py0 = ((float)r0 - 240.0f) * ad0 * rfy;
      const float px1 = ((float)c1 - 320.0f) * ad1 * rfx;
      const float py1 = ((float)r1 - 240.0f) * ad1 * rfy;
      const float v0 = (gz.x > 0.1f) ? 1.0f : 0.0f;   // branch-free select
      const float v1 = (gz.y > 0.1f) ? 1.0f : 0.0f;
      const float f0 = v0 * (su_px*px0 + su_py*py0 + su_1)
                          * (sv_px*px0 + sv_py*py0 + sv_d*dz.x + sv_1);
      const float f1 = v1 * (su_px*px1 + su_py*py1 + su_1)
                          * (sv_px*px1 + sv_py*py1 + sv_d*dz.y + sv_1);
      v2f bv; bv.x = f0; bv.y = f1;                // B: VGPR0=K(hi*2), VGPR1=K(hi*2+1)

      const unsigned char* mrow = smw + pl;
      #pragma unroll
      for (int t = 0; t < 8; ++t) {
        const unsigned int m =
            *(const unsigned short*)(mrow + (t * 16 + col) * ROWB);
        v2f av; av.x = (float)(m & 1u);            // A: VGPR0=K(hi*2)
                av.y = (float)((m >> 8) & 1u);     //    VGPR1=K(hi*2+1)
        acc[t] = __builtin_amdgcn_wmma_f32_16x16x4_f32(
            false, av, false, bv, (short)0, acc[t], false, false);
      }
    }
    __syncthreads();
  }

  // ---- block reduction: each wave dumps 8 tiles to its LDS slice, then 128 threads fold 4 slices
  float* red = (float*)smem;
  float* redw = red + wave * 2048;
  #pragma unroll
  for (int t = 0; t < 8; ++t) {
    #pragma unroll
    for (int rr = 0; rr < 8; ++rr)
      redw[(t * 16 + rr + hi * 8) * 16 + col] = acc[t][rr];
  }
  __syncthreads();
  for (int j = tid; j < 2048; j += MOM_THREADS) {
    float sum = red[j] + red[j + 2048] + red[j + 4096] + red[j + 6144];
    atomicAdd(Sg + j, sum);
  }
}

// ---------------------------------------------------------------- per-n 3x3 solve + scale
__global__ void solve_kernel(const float* __restrict__ Sg,
                             const float* __restrict__ d0,
                             const float* __restrict__ kmat,
                             const int* __restrict__ xp,
                             const int* __restrict__ yp,
                             float* __restrict__ fit,
                             int* __restrict__ okArr) {
  int n = blockIdx.x * blockDim.x + threadIdx.x;
  if (n >= NMASK) return;
  const float* s = Sg + n * 16;
  float m00=s[0], m01=s[1], m02=s[2], m11=s[3], m12=s[4], m22=s[5];
  float r0=s[6], r1=s[7], r2=s[8];
  float A00 =  (m11*m22 - m12*m12);
  float A01 = -(m01*m22 - m02*m12);
  float A02 =  (m01*m12 - m02*m11);
  float A11 =  (m00*m22 - m02*m02);
  float A12 = -(m00*m12 - m01*m02);
  float A22 =  (m00*m11 - m01*m01);
  float det = m00*A00 + m01*A01 + m02*A02;
  int ok = (fabsf(det) > 1e-8f) && (m22 >= 3.0f);   // m22 == cnt
  float a = 0.f, b = 0.f, dd = 0.f, scale = 0.f;
  if (ok) {
    float inv = 1.0f / det;
    a  = (A00*r0 + A01*r1 + A02*r2) * inv;
    b  = (A01*r0 + A11*r1 + A12*r2) * inv;
    dd = (A02*r0 + A12*r1 + A22*r2) * inv;
    float rfx = 1.0f / kmat[0], rfy = 1.0f / kmat[4];
    float sum = 0.f;
    for (int k = 0; k < KPTS; ++k) {
      int rr = xp[n * KPTS + k];
      int cc = yp[n * KPTS + k];
      float dz = d0[rr * WW + cc];
      float ad = fabsf(dz);
      float px = ((float)cc - 320.0f) * ad * rfx;
      float py = ((float)rr - 240.0f) * ad * rfy;
      sum += dz / (a * px + b * py + dd);
    }
    scale = sum * (1.0f / KPTS);
  }
  fit[n*4+0] = a; fit[n*4+1] = b; fit[n*4+2] = dd; fit[n*4+3] = scale;
  okArr[n] = ok;
}

// ---------------------------------------------------------------- ordered scatter via packed atomicMax
__global__ void scatter_kernel(const float* __restrict__ d0,
                               const float* __restrict__ kmat,
                               const int* __restrict__ xq,
                               const int* __restrict__ yq,
                               const float* __restrict__ fit,
                               const int* __restrict__ okArr,
                               unsigned long long* __restrict__ pk) {
  int idx = blockIdx.x * blockDim.x + threadIdx.x;
  if (idx >= NMASK * QQ) return;
  int n = idx >> 13;                     // Q = 8192
  if (!okArr[n]) return;
  int rr = xq[idx], cc = yq[idx];
  float dz = d0[rr * WW + cc];
  float ad = fabsf(dz);
  float rfx = 1.0f / kmat[0], rfy = 1.0f / kmat[4];
  float px = ((float)cc - 320.0f) * ad * rfx;
  float py = ((float)rr - 240.0f) * ad * rfy;
  float a = fit[n*4+0], b = fit[n*4+1], dd = fit[n*4+2], sc = fit[n*4+3];
  float val = (a * px + b * py + dd) * sc;
  unsigned long long packed =
      (((unsigned long long)(unsigned)(idx + 1)) << 32) |
      (unsigned long long)__float_as_uint(val);
  atomicMax(pk + (rr * WW + cc), packed);   // highest (n,q) wins == sequential scan order
}

// ---------------------------------------------------------------- finalize
__global__ void finalize_kernel(const unsigned long long* __restrict__ pk,
                                const float* __restrict__ d0,
                                float* __restrict__ out) {
  int i = blockIdx.x * blockDim.x + threadIdx.x;
  if (i >= NP) return;
  unsigned long long v = pk[i];
  out[i] = (v >> 32) ? __uint_as_float((unsigned)(v & 0xffffffffu)) : d0[i];
}

// ---------------------------------------------------------------- launch
extern "C" void kernel_launch(void* const* d_in, const int* in_sizes, int n_in,
                              void* d_out, int out_size, void* d_ws, size_t ws_size,
                              hipStream_t stream) {
  const float* d0            = (const float*)d_in[0];          // pred_depth (1,H,W)
  const unsigned char* gm    = (const unsigned char*)d_in[1];  // gt_masks (N,H,W) bool
  const float* gtd           = (const float*)d_in[2];          // gt_depth (1,H,W)
  const float* kmat          = (const float*)d_in[3];          // (3,3)
  const unsigned char* rm    = (const unsigned char*)d_in[4];  // random_mask (N,H,W) bool
  const int* xq              = (const int*)d_in[5];
  const int* yq              = (const int*)d_in[6];
  const int* xp              = (const int*)d_in[7];
  const int* yp              = (const int*)d_in[8];

  unsigned char* ws = (unsigned char*)d_ws;
  unsigned long long* pk = (unsigned long long*)ws;                 // NP * 8 B
  float* Sg  = (float*)(ws + (size_t)NP * 8);                       // 2048 * 4 B
  float* fit = (float*)(ws + (size_t)NP * 8 + 2048 * 4);            // 128 * 16 B
  int* okArr = (int*)  (ws + (size_t)NP * 8 + 2048 * 4 + 128 * 16); // 128 * 4 B

  init_kernel<<<(NP + 255) / 256, 256, 0, stream>>>(pk, Sg);
  moment_kernel<<<MOM_BLOCKS, MOM_THREADS, 0, stream>>>(d0, gm, rm, gtd, kmat, Sg);
  solve_kernel<<<1, 128, 0, stream>>>(Sg, d0, kmat, xp, yp, fit, okArr);
  scatter_kernel<<<(NMASK * QQ) / 256, 256, 0, stream>>>(d0, kmat, xq, yq, fit, okArr, pk);
  finalize_kernel<<<(NP + 255) / 256, 256, 0, stream>>>(pk, d0, (float*)d_out);
}